// EntropyLoss_89893665505443
// MI455X (gfx1250) — compile-verified
//
#include <hip/hip_runtime.h>
#include <stdint.h>

#define NBINS 256
#define BLOCK 256
#define NWAVES (BLOCK / 32)
#define VEC 4
#define CHUNK (BLOCK * VEC) /* 1024 floats (4 KB) staged per chunk per block */
#define HEPS 1e-8f

// ---------------------------------------------------------------------------
// CDNA5 async global->LDS copy: each lane moves 16 bytes; tracked by ASYNCcnt.
// VDST = VGPR holding LDS byte offset, VADDR = 64-bit global address (GV mode).
// ---------------------------------------------------------------------------
__device__ __forceinline__ void async_copy16(const float* gptr, const void* lptr) {
  unsigned l = (unsigned)(unsigned long long)lptr;   // low 32 bits = LDS offset
  unsigned long long g = (unsigned long long)gptr;
  asm volatile("global_load_async_to_lds_b128 %0, %1, off"
               :: "v"(l), "v"(g) : "memory");
}

// Faithful to torch.histc semantics used by the reference:
// width = 2/256 = 2^-7, so (x+1)*128 is bit-exact vs (x-lo)/width.
// x outside [-1,1] (and NaN) ignored; x == 1.0 goes to the last bin.
__device__ __forceinline__ void bump(unsigned* h, float x) {
  if (x >= -1.0f && x <= 1.0f) {
    int idx = (int)floorf((x + 1.0f) * 128.0f);
    idx = idx > (NBINS - 1) ? (NBINS - 1) : idx;
    atomicAdd(&h[idx], 1u);   // ds_add_u32 (wave-private copy -> low contention)
  }
}

__global__ void __launch_bounds__(512) zero_hist_kernel(unsigned* hist) {
  if (threadIdx.x < 2 * NBINS) hist[threadIdx.x] = 0u;
}

__global__ void __launch_bounds__(BLOCK)
hist_kernel(const float* __restrict__ pred, const float* __restrict__ gt,
            unsigned* __restrict__ hist, int n) {
  __shared__ __align__(16) float stage[2][CHUNK];   // 8 KB double buffer
  __shared__ unsigned lhist[NWAVES][NBINS];         // 8 KB per-wave histograms

  const int tid = threadIdx.x;
  for (int i = tid; i < NWAVES * NBINS; i += BLOCK) (&lhist[0][0])[i] = 0u;
  __syncthreads();

  const float* __restrict__ src = (blockIdx.y == 0) ? pred : gt;
  unsigned* __restrict__ ghist = hist + blockIdx.y * NBINS;
  unsigned* myh = lhist[tid >> 5];

  const long long stride = (long long)gridDim.x * CHUNK;
  const long long base0  = (long long)blockIdx.x * CHUNK;
  const int lane4 = tid * VEC;

  // Prologue: stage the first chunk asynchronously.
  if (base0 < n) {
    long long off = base0 + lane4;
    if (off + VEC <= (long long)n) async_copy16(src + off, &stage[0][lane4]);
  }

  int buf = 0;
  for (long long base = base0; base < (long long)n; base += stride, buf ^= 1) {
    const long long nbase = base + stride;
    // Wave-uniform: did this wave actually issue a copy for the next chunk?
    const bool issued_next = (nbase < (long long)n) &&
                             (nbase + VEC <= (long long)n); // lane-0 condition
    if (nbase < (long long)n) {
      long long noff = nbase + lane4;
      if (noff + VEC <= (long long)n)
        async_copy16(src + noff, &stage[buf ^ 1][lane4]);
    }
    // Async loads complete in order: cnt<=1 ==> current buffer has landed.
    if (issued_next) asm volatile("s_wait_asynccnt 0x1" ::: "memory");
    else             asm volatile("s_wait_asynccnt 0x0" ::: "memory");

    long long off = base + lane4;
    if (off + VEC <= (long long)n) {
      float4 v = *(const float4*)&stage[buf][lane4];   // ds_load_b128
      bump(myh, v.x); bump(myh, v.y); bump(myh, v.z); bump(myh, v.w);
    } else {
      // Tail (never taken for n = 33.5M, kept for safety) — direct global reads.
      for (int k = 0; k < VEC; ++k)
        if (off + k < (long long)n) bump(myh, src[off + k]);
    }
  }

  __syncthreads();
  // Merge the 8 wave-private histograms; one global atomic per bin per block.
  unsigned total = 0;
#pragma unroll
  for (int w = 0; w < NWAVES; ++w) total += lhist[w][tid];
  if (total) atomicAdd(&ghist[tid], total);
}

__device__ float block_reduce_sum(float v, float* red) {
  int t = threadIdx.x;
  red[t] = v;
  __syncthreads();
  for (int s = BLOCK / 2; s > 0; s >>= 1) {
    if (t < s) red[t] += red[t + s];
    __syncthreads();
  }
  float r = red[0];
  __syncthreads();
  return r;
}

__global__ void __launch_bounds__(BLOCK)
entropy_kernel(const unsigned* __restrict__ hist, float* __restrict__ out) {
  __shared__ float red[BLOCK];
  int t = threadIdx.x;
  float h0 = (float)hist[t];
  float h1 = (float)hist[NBINS + t];
  float s0 = block_reduce_sum(h0, red);
  float s1 = block_reduce_sum(h1, red);
  float p0 = h0 / s0 + HEPS;
  float p1 = h1 / s1 + HEPS;
  float e0 = -block_reduce_sum(p0 * logf(p0), red);
  float e1 = -block_reduce_sum(p1 * logf(p1), red);
  if (t == 0) out[0] = fabsf(e0 - e1);
}

extern "C" void kernel_launch(void* const* d_in, const int* in_sizes, int n_in,
                              void* d_out, int out_size, void* d_ws, size_t ws_size,
                              hipStream_t stream) {
  const float* pred = (const float*)d_in[0];
  const float* gt   = (const float*)d_in[1];
  int n = in_sizes[0];
  unsigned* hist = (unsigned*)d_ws;   // 2 x 256 u32 = 2 KB scratch
  float* out = (float*)d_out;

  zero_hist_kernel<<<1, 512, 0, stream>>>(hist);

  int chunks = (n + CHUNK - 1) / CHUNK;
  int gx = chunks < 4096 ? (chunks > 0 ? chunks : 1) : 4096;
  hist_kernel<<<dim3((unsigned)gx, 2), BLOCK, 0, stream>>>(pred, gt, hist, n);

  entropy_kernel<<<1, BLOCK, 0, stream>>>(hist, out);
}